// GeoRegionSampler_23021024706570
// MI455X (gfx1250) — compile-verified
//
#include <hip/hip_runtime.h>
#include <stdint.h>

#define Bdim 8
#define Cdim 64
#define Hdim 512
#define Wdim 512
#define CPB  8          // channels staged per block
#define NTHREADS 256    // 8 waves (wave32)

typedef float        v4f   __attribute__((ext_vector_type(4)));
typedef unsigned int u32x4 __attribute__((ext_vector_type(4)));
typedef int          i32x4 __attribute__((ext_vector_type(4)));
typedef int          i32x8 __attribute__((ext_vector_type(8)));

#if __has_builtin(__builtin_amdgcn_tensor_load_to_lds)
#define HAVE_TDM 1
#else
#define HAVE_TDM 0
#endif

__global__ __launch_bounds__(NTHREADS)
void geo_region_sampler_tdm(const float* __restrict__ img,
                            const int*   __restrict__ boxes,
                            float*       __restrict__ out)
{
  // [channel][row(2)][x(cw, packed)] — TDM packs the tile contiguously in LDS.
  __shared__ float lds[CPB * 2 * Wdim];   // 32 KB worst case (cw <= 511)

  const int y   = blockIdx.x;          // output row
  const int b   = blockIdx.y;          // batch
  const int c0  = blockIdx.z * CPB;    // first channel of this block
  const int tid = (int)threadIdx.x;

  // ---- box (block-uniform; force scalar for the descriptor path) ----
  int x1 = __builtin_amdgcn_readfirstlane(boxes[b*4 + 0]);
  int y1 = __builtin_amdgcn_readfirstlane(boxes[b*4 + 1]);
  int x2 = __builtin_amdgcn_readfirstlane(boxes[b*4 + 2]);
  int y2 = __builtin_amdgcn_readfirstlane(boxes[b*4 + 3]);
  x1 = min(max(x1, 0), Wdim - 1);  x2 = min(max(x2, 0), Wdim - 1);
  y1 = min(max(y1, 0), Hdim - 1);  y2 = min(max(y2, 0), Hdim - 1);
  const int   ch  = y2 - y1;
  const int   cw  = x2 - x1;
  const float chf = (float)ch;
  const float cwf = (float)cw;

  // ---- row params (uniform across the block) ----
  float sy = ((float)y + 0.5f) * (chf * (1.0f / (float)Hdim)) - 0.5f;
  sy = fminf(fmaxf(sy, 0.0f), chf - 1.0f);
  const int   y0i = (int)sy;                 // sy >= 0 -> trunc == floor
  const int   y1n = min(y0i + 1, ch - 1);
  const float wy  = sy - (float)y0i;
  const int   r0  = y1 + y0i;
  const int   r1  = y1 + y1n;

#if HAVE_TDM
  // ---- Tensor DMA: stage (cw x 2rows x CPBchannels) fp32 tile into LDS ----
  {
    const uint64_t gaddr = (uint64_t)(uintptr_t)img +
        4ull * ((uint64_t)((b*Cdim + c0)*Hdim + r0) * (uint64_t)Wdim + (uint64_t)x1);
    const uint32_t ldsa = (uint32_t)(uintptr_t)(&lds[0]);
    const uint32_t dim0 = (uint32_t)cw;                      // elements per row
    const uint32_t str0 = (uint32_t)((r1 - r0) * Wdim);      // dim-1 step (elems)
    const uint64_t str1 = (uint64_t)Hdim * (uint64_t)Wdim;   // dim-2 step (elems)

    u32x4 g0;
    g0[0] = 1u;                                              // count=1, no gather
    g0[1] = ldsa;                                            // lds_addr (bytes)
    g0[2] = (uint32_t)(gaddr & 0xFFFFFFFFull);               // global_addr[31:0]
    g0[3] = (uint32_t)((gaddr >> 32) & 0x01FFFFFFull)        // global_addr[56:32]
          | (2u << 30);                                      // type=2 ("image")

    i32x8 g1;
    g1[0] = (int)(2u << 16);                                 // data_size=2 (4B)
    g1[1] = (int)((dim0 & 0xFFFFu) << 16);                   // tensor_dim0[15:0]
    g1[2] = (int)((dim0 >> 16) | (2u << 16));                // dim0[31:16] | tensor_dim1=2
    g1[3] = (int)((dim0 & 0xFFFFu) << 16);                   // tile_dim0 = cw
    g1[4] = (int)(2u | ((unsigned)CPB << 16));               // tile_dim1=2, tile_dim2=CPB
    g1[5] = (int)str0;                                       // tensor_dim0_stride lo32
    g1[6] = (int)(((uint32_t)(str1 & 0xFFFFull)) << 16);     // str0 hi16=0 | str1[15:0]
    g1[7] = (int)((uint32_t)((str1 >> 16) & 0xFFFFFFFFull)); // str1[47:16]

    i32x4 g2; g2[0] = CPB; g2[1] = 0; g2[2] = 0; g2[3] = 0;  // tensor_dim2=CPB
    i32x4 g3 = {};

    if (tid < 32) {                                          // wave 0 issues the DMA
#if defined(__clang_major__) && (__clang_major__ >= 23)
      i32x8 gz = {};
      __builtin_amdgcn_tensor_load_to_lds(g0, g1, g2, g3, gz, 0);
#else
      __builtin_amdgcn_tensor_load_to_lds(g0, g1, g2, g3, 0);
#endif
    }
  }
#endif

  // ---- per-column params, computed while the DMA is in flight ----
  const int xg   = (tid & 127) * 4;   // 4 consecutive output x per thread
  const int csub = tid >> 7;          // 0/1: which channel of the pair
  int   xi0[4], xi1[4];
  float wxv[4];
  #pragma unroll
  for (int j = 0; j < 4; ++j) {
    float sx = ((float)(xg + j) + 0.5f) * (cwf * (1.0f / (float)Wdim)) - 0.5f;
    sx = fminf(fmaxf(sx, 0.0f), cwf - 1.0f);
    const int x0i = (int)sx;          // sx >= 0 -> trunc == floor
    xi0[j] = x0i;
    xi1[j] = min(x0i + 1, cw - 1);
    wxv[j] = sx - (float)x0i;
  }

#if HAVE_TDM
#if __has_builtin(__builtin_amdgcn_s_wait_tensorcnt)
  if (tid < 32) __builtin_amdgcn_s_wait_tensorcnt(0);
#endif
#else
  // fallback: cooperative staging through VGPRs
  for (int i = tid; i < CPB * 2 * cw; i += NTHREADS) {
    const int cl  = i / (2 * cw);
    const int rem = i - cl * (2 * cw);
    const int row = rem / cw;
    const int xi  = rem - row * cw;
    const int r   = row ? r1 : r0;
    lds[i] = img[(size_t)((b*Cdim + c0 + cl)*Hdim + r) * Wdim + (size_t)(x1 + xi)];
  }
#endif
  __syncthreads();

  // ---- bilinear blend from LDS, streaming float4 stores ----
  const float wy1 = 1.0f - wy;
  #pragma unroll
  for (int cc = 0; cc < CPB / 2; ++cc) {
    const int cl = cc * 2 + csub;
    const float* __restrict__ l0 = &lds[(cl*2 + 0) * cw];   // row r0 segment
    const float* __restrict__ l1 = &lds[(cl*2 + 1) * cw];   // row r1 segment
    v4f o;
    #pragma unroll
    for (int j = 0; j < 4; ++j) {
      const float wx  = wxv[j];
      const float wx1 = 1.0f - wx;
      const float v00 = l0[xi0[j]], v01 = l0[xi1[j]];
      const float v10 = l1[xi0[j]], v11 = l1[xi1[j]];
      const float top = v00 * wx1 + v01 * wx;
      const float bot = v10 * wx1 + v11 * wx;
      o[j] = top * wy1 + bot * wy;
    }
    const size_t oidx = ((size_t)((b*Cdim + c0 + cl)*Hdim + y)) * Wdim + (size_t)xg;
    __builtin_nontemporal_store(o, (v4f*)(out + oidx));     // output never re-read
  }
}

extern "C" void kernel_launch(void* const* d_in, const int* in_sizes, int n_in,
                              void* d_out, int out_size, void* d_ws, size_t ws_size,
                              hipStream_t stream) {
  (void)in_sizes; (void)n_in; (void)out_size; (void)d_ws; (void)ws_size;
  const float* img   = (const float*)d_in[0];
  const int*   boxes = (const int*)d_in[1];
  float*       out   = (float*)d_out;
  dim3 grid(Hdim, Bdim, Cdim / CPB);   // (y, batch, channel-block)
  geo_region_sampler_tdm<<<grid, NTHREADS, 0, stream>>>(img, boxes, out);
}